// SinkhornLoss_4810363373030
// MI455X (gfx1250) — compile-verified
//
#include <hip/hip_runtime.h>
#include <math.h>

// ---------------- problem constants ----------------
#define NDIM 8192          // N == M
#define DDIM 128
#define EPS  0.001f
#define ITERS 100
#define SPLIT 64           // row-chunks for deterministic column matvec
#define RPC  (NDIM / SPLIT)

typedef __attribute__((ext_vector_type(2))) float v2f;
typedef __attribute__((ext_vector_type(8))) float v8f;

// ---------------- init ----------------
__global__ void init_kernel(float* u, float* v, int* done, unsigned* wmax_bits) {
    int i = blockIdx.x * blockDim.x + threadIdx.x;
    if (i < NDIM) {
        u[i] = 1.0f / (float)NDIM;
        v[i] = 1.0f / (float)NDIM;
    }
    if (i == 0) { *done = 0; *wmax_bits = 0u; }
}

// ---------------- row squared norms (one block of 128 per row) ----------------
__global__ __launch_bounds__(128) void sqnorm_kernel(const float* __restrict__ a,
                                                     float* __restrict__ out) {
    __shared__ float red[128];
    const int row = blockIdx.x;
    float t = a[(size_t)row * DDIM + threadIdx.x];
    red[threadIdx.x] = t * t;
    __syncthreads();
    for (int s = 64; s > 0; s >>= 1) {
        if (threadIdx.x < s) red[threadIdx.x] += red[threadIdx.x + s];
        __syncthreads();
    }
    if (threadIdx.x == 0) out[row] = red[0];
}

// ---------------- WMMA f32 GEMM -> W = sqrt(max(x2+y2-2*x.y, 0)), track max ----
// Macrotile 64x32, 8 waves, each wave one 16x16 WMMA tile (K-loop of 32 x K=4).
#define MT_R 64
#define MT_C 32
__global__ __launch_bounds__(256) void gemm_w_kernel(const float* __restrict__ x,
                                                     const float* __restrict__ y,
                                                     const float* __restrict__ x2,
                                                     const float* __restrict__ y2,
                                                     float* __restrict__ W,
                                                     unsigned* __restrict__ wmax_bits) {
    __shared__ float xs[MT_R][DDIM + 2];   // pad 2 floats: keeps 8B align, spreads banks
    __shared__ float ys[MT_C][DDIM + 2];
    __shared__ float red[256];

    const int tid = threadIdx.x;
    const int r0 = blockIdx.x * MT_R;
    const int c0 = blockIdx.y * MT_C;

    for (int idx = tid; idx < MT_R * DDIM; idx += 256) {
        int rr = idx >> 7, cc = idx & 127;
        xs[rr][cc] = x[(size_t)(r0 + rr) * DDIM + cc];
    }
    for (int idx = tid; idx < MT_C * DDIM; idx += 256) {
        int rr = idx >> 7, cc = idx & 127;
        ys[rr][cc] = y[(size_t)(c0 + rr) * DDIM + cc];
    }
    __syncthreads();

    const int wave = tid >> 5;
    const int lane = tid & 31;
    const int mBase = (wave >> 1) * 16;      // 4 tile-rows
    const int nBase = (wave & 1) * 16;       // 2 tile-cols
    const int l15 = lane & 15;
    const int khalf = (lane >> 4) << 1;      // 0 for lanes 0-15, 2 for lanes 16-31

    // A 16x4 f32 frag: lane m holds (K=khalf, K=khalf+1); B 4x16 mirrors with N=lane.
    v8f acc = {};
    for (int kb = 0; kb < DDIM; kb += 4) {
        v2f a, b;
        a.x = xs[mBase + l15][kb + khalf];
        a.y = xs[mBase + l15][kb + khalf + 1];
        b.x = ys[nBase + l15][kb + khalf];
        b.y = ys[nBase + l15][kb + khalf + 1];
        acc = __builtin_amdgcn_wmma_f32_16x16x4_f32(false, a, false, b,
                                                    (short)0, acc, false, false);
    }

    // C/D layout: VGPR p -> M = p + 8*(lane>=16), N = lane&15
    const int col = c0 + nBase + l15;
    const float yc = y2[col];
    float lmax = 0.0f;
#pragma unroll
    for (int p = 0; p < 8; ++p) {
        int row = r0 + mBase + p + ((lane >> 4) << 3);
        float w = x2[row] + yc - 2.0f * acc[p];
        w = sqrtf(fmaxf(w, 0.0f));
        W[(size_t)row * NDIM + col] = w;
        lmax = fmaxf(lmax, w);
    }
    red[tid] = lmax;
    __syncthreads();
    for (int s = 128; s > 0; s >>= 1) {
        if (tid < s) red[tid] = fmaxf(red[tid], red[tid + s]);
        __syncthreads();
    }
    if (tid == 0) atomicMax(wmax_bits, __float_as_uint(red[0]));  // W>=0: bit order == float order
}

// ---------------- K = exp(-eps * (W/wmax) / D), in place ----------------
__global__ void expmap_kernel(float* __restrict__ W, const unsigned* __restrict__ wmax_bits,
                              size_t n) {
    const float wmax = __uint_as_float(*wmax_bits);
    size_t i = (size_t)blockIdx.x * blockDim.x + threadIdx.x;
    size_t stride = (size_t)gridDim.x * blockDim.x;
    for (; i < n; i += stride) {
        float wn = W[i] / wmax;                 // match reference op order
        W[i] = expf((-EPS * wn) / (float)DDIM);
    }
}

// ---------------- u = 1/(K@v)  (one block per row, deterministic LDS tree) ----
__global__ __launch_bounds__(256) void matvec_row_kernel(const float* __restrict__ K,
                                                         const float* __restrict__ vin,
                                                         float* __restrict__ uout,
                                                         const int* __restrict__ done,
                                                         int use_done, int recip) {
    if (use_done && *done) return;
    __shared__ float red[256];
    const int row = blockIdx.x;
    const float4* K4 = (const float4*)(K + (size_t)row * NDIM);
    const float4* v4 = (const float4*)vin;
    float acc = 0.0f;
    for (int j = threadIdx.x; j < NDIM / 4; j += 256) {
        float4 k = K4[j], vv = v4[j];
        acc += k.x * vv.x + k.y * vv.y + k.z * vv.z + k.w * vv.w;
    }
    red[threadIdx.x] = acc;
    __syncthreads();
    for (int s = 128; s > 0; s >>= 1) {
        if (threadIdx.x < s) red[threadIdx.x] += red[threadIdx.x + s];
        __syncthreads();
    }
    if (threadIdx.x == 0) uout[row] = recip ? 1.0f / red[0] : red[0];
}

// ---------------- v partials: vpart[s][:] = sum over row-chunk s of K^T u ------
__global__ __launch_bounds__(256) void matvec_col_part_kernel(const float* __restrict__ K,
                                                              const float* __restrict__ uin,
                                                              float* __restrict__ vpart,
                                                              const int* __restrict__ done) {
    if (*done) return;
    const int colv = blockIdx.x * 256 + threadIdx.x;   // float4 column index (0..2047)
    const int s = blockIdx.y;
    const int i0 = s * RPC;
    const float4* K4 = (const float4*)K;
    float4 acc = {0.f, 0.f, 0.f, 0.f};
    for (int i = 0; i < RPC; ++i) {
        float4 k = K4[(size_t)(i0 + i) * (NDIM / 4) + colv];
        float us = uin[i0 + i];
        acc.x += k.x * us; acc.y += k.y * us; acc.z += k.z * us; acc.w += k.w * us;
    }
    ((float4*)vpart)[(size_t)s * (NDIM / 4) + colv] = acc;
}

__global__ __launch_bounds__(256) void matvec_col_reduce_kernel(const float* __restrict__ vpart,
                                                                float* __restrict__ vout,
                                                                const int* __restrict__ done) {
    if (*done) return;
    const int colv = blockIdx.x * 256 + threadIdx.x;   // float4 column index
    const float4* p4 = (const float4*)vpart;
    float4 acc = {0.f, 0.f, 0.f, 0.f};
    for (int s = 0; s < SPLIT; ++s) {                  // fixed order -> deterministic
        float4 k = p4[(size_t)s * (NDIM / 4) + colv];
        acc.x += k.x; acc.y += k.y; acc.z += k.z; acc.w += k.w;
    }
    float4 r = {1.0f / acc.x, 1.0f / acc.y, 1.0f / acc.z, 1.0f / acc.w};
    ((float4*)vout)[colv] = r;
}

// ---------------- helpers: copy, convergence, final dot ----------------
__global__ void copy_kernel(const float* __restrict__ src, float* __restrict__ dst,
                            const int* __restrict__ done) {
    if (*done) return;
    int i = blockIdx.x * blockDim.x + threadIdx.x;
    if (i < NDIM) dst[i] = src[i];
}

__global__ __launch_bounds__(1024) void conv_kernel(const float* __restrict__ u,
                                                    const float* __restrict__ uprev,
                                                    int* __restrict__ done) {
    if (*done) return;
    __shared__ float red[1024];
    float acc = 0.0f;
    for (int j = threadIdx.x; j < NDIM; j += 1024) acc += fabsf(u[j] - uprev[j]);
    red[threadIdx.x] = acc;
    __syncthreads();
    for (int s = 512; s > 0; s >>= 1) {
        if (threadIdx.x < s) red[threadIdx.x] += red[threadIdx.x + s];
        __syncthreads();
    }
    if (threadIdx.x == 0 && red[0] < EPS) *done = 1;
}

__global__ __launch_bounds__(1024) void final_dot_kernel(const float* __restrict__ u,
                                                         const float* __restrict__ t,
                                                         float* __restrict__ out) {
    __shared__ float red[1024];
    float acc = 0.0f;
    for (int j = threadIdx.x; j < NDIM; j += 1024) acc += u[j] * t[j];
    red[threadIdx.x] = acc;
    __syncthreads();
    for (int s = 512; s > 0; s >>= 1) {
        if (threadIdx.x < s) red[threadIdx.x] += red[threadIdx.x + s];
        __syncthreads();
    }
    if (threadIdx.x == 0) out[0] = red[0] / (float)NDIM;
}

// ---------------- launch ----------------
extern "C" void kernel_launch(void* const* d_in, const int* in_sizes, int n_in,
                              void* d_out, int out_size, void* d_ws, size_t ws_size,
                              hipStream_t stream) {
    const float* x = (const float*)d_in[0];
    const float* y = (const float*)d_in[1];
    float* out = (float*)d_out;

    char* ws = (char*)d_ws;
    size_t off = 0;
    float* K = (float*)(ws + off);      off += (size_t)NDIM * NDIM * sizeof(float); // 256 MB
    float* u = (float*)(ws + off);      off += NDIM * sizeof(float);
    float* uprev = (float*)(ws + off);  off += NDIM * sizeof(float);
    float* v = (float*)(ws + off);      off += NDIM * sizeof(float);
    float* t = (float*)(ws + off);      off += NDIM * sizeof(float);
    float* x2 = (float*)(ws + off);     off += NDIM * sizeof(float);
    float* y2 = (float*)(ws + off);     off += NDIM * sizeof(float);
    float* vpart = (float*)(ws + off);  off += (size_t)SPLIT * NDIM * sizeof(float); // 2 MB
    int* done = (int*)(ws + off);       off += sizeof(int);
    unsigned* wmaxb = (unsigned*)(ws + off); off += sizeof(unsigned);
    (void)ws_size; (void)in_sizes; (void)n_in; (void)out_size;

    init_kernel<<<NDIM / 256, 256, 0, stream>>>(u, v, done, wmaxb);
    sqnorm_kernel<<<NDIM, 128, 0, stream>>>(x, x2);
    sqnorm_kernel<<<NDIM, 128, 0, stream>>>(y, y2);

    dim3 g(NDIM / MT_R, NDIM / MT_C);
    gemm_w_kernel<<<g, 256, 0, stream>>>(x, y, x2, y2, K, wmaxb);
    expmap_kernel<<<8192, 256, 0, stream>>>(K, wmaxb, (size_t)NDIM * NDIM);

    for (int it = 0; it < ITERS; ++it) {
        copy_kernel<<<NDIM / 256, 256, 0, stream>>>(u, uprev, done);
        matvec_row_kernel<<<NDIM, 256, 0, stream>>>(K, v, u, done, 1, 1);
        matvec_col_part_kernel<<<dim3((NDIM / 4) / 256, SPLIT), 256, 0, stream>>>(K, u, vpart, done);
        matvec_col_reduce_kernel<<<(NDIM / 4) / 256, 256, 0, stream>>>(vpart, v, done);
        conv_kernel<<<1, 1024, 0, stream>>>(u, uprev, done);
    }

    // sinkhorn_div = sum(u * (K @ v_final)) / n  (always executed)
    matvec_row_kernel<<<NDIM, 256, 0, stream>>>(K, v, t, done, 0, 0);
    final_dot_kernel<<<1, 1024, 0, stream>>>(u, t, out);
}